// waspShadeRenderer_10857677325055
// MI455X (gfx1250) — compile-verified
//
#include <hip/hip_runtime.h>
#include <hip/hip_bf16.h>

typedef __attribute__((ext_vector_type(2))) float v2f;
typedef __attribute__((ext_vector_type(8))) float v8f;

#define SH_C1 0.429043f
#define SH_C2 0.511664f
#define SH_C3 0.743152f
#define SH_C4 0.886227f
#define SH_C5 0.247708f

#define WH (512 * 512)          // pixels per batch plane
#define PIX_PER_BLOCK 4096      // WH / 64 blocks
#define THREADS 256             // 8 waves
#define ITERS (PIX_PER_BLOCK / THREADS)  // 16

// s(p) = n4^T M n4 via V_WMMA_F32_16X16X4_F32:
//   A (16x4) = rows 0..3 = M, rows 8..11 = M (duplicate), rest 0
//   B (4x16) = 16 pixels' [x y z 1] as columns
// D layout: VGPR v, lanes 0-15 = row v, lanes 16-31 = row v+8. So:
//   group A (cols = pixels 0..15):  lane p      gets (M n_p)_v      in d[v] (rows 0-3)
//   group B (cols = pixels 16..31): lane n+16   gets (M n_{n+16})_v in d[v] (rows 8-11)
// -> every lane finishes its own pixel's dot product lane-locally; zero
//    post-WMMA cross-lane traffic.
__global__ __launch_bounds__(THREADS) void sh_irradiance_wmma(
    const float* __restrict__ light,    // [64, 9]
    const float* __restrict__ normals,  // [64, 3, 512, 512]
    float* __restrict__ out)            // [64, 1, 512, 512]
{
    const int b = blockIdx.y;

    // ---- uniform (SGPR) per-batch irradiance matrix M ----
    const float* L = light + b * 9;
    const float L0 = L[0], L1 = L[1], L2 = L[2], L3 = L[3], L4 = L[4];
    const float L5 = L[5], L6 = L[6], L7 = L[7], L8 = L[8];

    const float M00 = SH_C1 * L8, M01 = SH_C1 * L4, M02 = SH_C1 * L7, M03 = SH_C2 * L3;
    const float M10 = M01, M11 = -SH_C1 * L8, M12 = SH_C1 * L5, M13 = SH_C2 * L1;
    const float M20 = M02, M21 = M12, M22 = SH_C3 * L6, M23 = SH_C2 * L2;
    const float M30 = M03, M31 = M13, M32 = M23, M33 = SH_C4 * L0 - SH_C5 * L6;

    const int lane = threadIdx.x & 31;
    const int m    = lane & 15;       // A-matrix row slot for this lane
    const bool hi  = lane >= 16;      // hi half: A cols 2,3 / B rows 2,3

    // ---- A operand (16x4 f32): lane m holds A[m, hi?2:0] in elem0 and
    //      A[m, hi?3:1] in elem1. Rows 0-3 and 8-11 carry M (row = m&3). ----
    float a0 = 0.0f, a1 = 0.0f;
    if ((m & 4) == 0) {               // m in {0..3, 8..11}
        const int r = m & 3;
        if (r == 0)      { a0 = hi ? M02 : M00; a1 = hi ? M03 : M01; }
        else if (r == 1) { a0 = hi ? M12 : M10; a1 = hi ? M13 : M11; }
        else if (r == 2) { a0 = hi ? M22 : M20; a1 = hi ? M23 : M21; }
        else             { a0 = hi ? M32 : M30; a1 = hi ? M33 : M31; }
    }
    const v2f A = { a0, a1 };
    const v8f cz = { 0.f, 0.f, 0.f, 0.f, 0.f, 0.f, 0.f, 0.f };

    // ---- streaming pointers (zero reuse -> non-temporal) ----
    const size_t plane = (size_t)b * 3 * WH;
    const float* nx = normals + plane;
    const float* ny = nx + WH;
    const float* nz = ny + WH;
    float* o = out + (size_t)b * WH;

    const int base = blockIdx.x * PIX_PER_BLOCK + threadIdx.x;

#pragma unroll 4
    for (int i = 0; i < ITERS; ++i) {
        const int idx = base + i * THREADS;

        // one pixel per lane: 32 consecutive pixels per wave, fully coalesced
        const float x = __builtin_nontemporal_load(nx + idx);
        const float y = __builtin_nontemporal_load(ny + idx);
        const float z = __builtin_nontemporal_load(nz + idx);

        // minimal cross-half traffic: 3 shuffles to build the two B operands
        const float zx = __shfl_xor(z, 16, 32);
        const float xx = __shfl_xor(x, 16, 32);
        const float yx = __shfl_xor(y, 16, 32);

        // Group A: columns = pixels 0..15 (B[0,n]=x_n, B[1,n]=y_n, B[2,n]=z_n, B[3,n]=1)
        const v2f Ba = { hi ? zx : x, hi ? 1.0f : y };
        // Group B: columns = pixels 16..31
        const v2f Bb = { hi ? z : xx, hi ? 1.0f : yx };

        v8f dA = __builtin_amdgcn_wmma_f32_16x16x4_f32(
            false, A, false, Ba, (short)0, cz, false, false);
        v8f dB = __builtin_amdgcn_wmma_f32_16x16x4_f32(
            false, A, false, Bb, (short)0, cz, false, false);

        // Lane-local finish: lo lanes use dA rows 0-3 (their pixel),
        // hi lanes use dB rows 8-11 (their pixel, via the duplicated M rows).
        const float g0 = hi ? dB[0] : dA[0];
        const float g1 = hi ? dB[1] : dA[1];
        const float g2 = hi ? dB[2] : dA[2];
        const float g3 = hi ? dB[3] : dA[3];
        const float s  = g0 * x + g1 * y + g2 * z + g3;

        __builtin_nontemporal_store(s, o + idx);
    }
}

extern "C" void kernel_launch(void* const* d_in, const int* in_sizes, int n_in,
                              void* d_out, int out_size, void* d_ws, size_t ws_size,
                              hipStream_t stream) {
    const float* light   = (const float*)d_in[0];   // [64, 9]
    const float* normals = (const float*)d_in[1];   // [64, 3, 512, 512]
    float* out = (float*)d_out;                     // [64, 1, 512, 512]

    dim3 grid(WH / PIX_PER_BLOCK, 64);  // (64 pixel chunks, 64 batches)
    dim3 block(THREADS);
    sh_irradiance_wmma<<<grid, block, 0, stream>>>(light, normals, out);
}